// QBottleneck_18287970747078
// MI455X (gfx1250) — compile-verified
//
#include <hip/hip_runtime.h>

// QBottleneck: 4-qubit PennyLane circuit, analytically reduced.
//
// Per sample i (x layout [B*C, 4] contiguous), with theta_q = x[q] + params[q]
// and t_q = cos(theta_q):
//   out[0] = t1*t2*t3
//   out[1] = t0*t1
//   out[2] = t0*t1*t2
//   out[3] = t0*t1*t2*t3
//
// Derivation: RY(x)RY(p)|0> = RY(x+p)|0> keeps a product state; the CNOT ring
// is a classical basis permutation (b0'=b1^b2^b3, b1'=b0^b1, b2'=b0^b1^b2,
// b3'=b0^b1^b2^b3); <Z_i> = prod over the XOR set of (cos^2 - sin^2)(th/2)
// = prod cos(th). Valid for the single entangler layer in setup_inputs
// (params shape [1,4]).
//
// Bandwidth-bound: 32 MB total traffic -> ~1.4 us floor at 23.3 TB/s.
// Single-pass streaming, so use non-temporal (TH_NT) b128 loads/stores.

typedef float v4f __attribute__((ext_vector_type(4)));

__global__ __launch_bounds__(256) void qbottleneck_kernel(
    const v4f* __restrict__ x,      // [n] float4 (one sample of 4 angles)
    const float* __restrict__ prm,  // [4] uniform -> scalar loads
    v4f* __restrict__ out,          // [n] float4
    int n)
{
    int i = blockIdx.x * 256 + threadIdx.x;
    if (i >= n) return;

    // Uniform address: compiler emits s_load_b128 for these.
    const float p0 = prm[0];
    const float p1 = prm[1];
    const float p2 = prm[2];
    const float p3 = prm[3];

    // Streamed, read-once: non-temporal 128-bit load.
    v4f v = __builtin_nontemporal_load(&x[i]);

    // v_cos_f32 (hardware transcendental) via native cosf.
    const float t0 = __cosf(v.x + p0);
    const float t1 = __cosf(v.y + p1);
    const float t2 = __cosf(v.z + p2);
    const float t3 = __cosf(v.w + p3);

    const float t01 = t0 * t1;
    const float t23 = t2 * t3;

    v4f r;
    r.x = t1 * t23;        // <Z0> = t1 t2 t3
    r.y = t01;             // <Z1> = t0 t1
    r.z = t01 * t2;        // <Z2> = t0 t1 t2
    r.w = t01 * t23;       // <Z3> = t0 t1 t2 t3

    // Written-once, never re-read by this kernel: non-temporal 128-bit store.
    __builtin_nontemporal_store(r, &out[i]);
}

extern "C" void kernel_launch(void* const* d_in, const int* in_sizes, int n_in,
                              void* d_out, int out_size, void* d_ws, size_t ws_size,
                              hipStream_t stream) {
    const v4f*   x   = (const v4f*)d_in[0];    // [2048*512, 4] f32
    const float* prm = (const float*)d_in[1];  // [1, 4] f32
    v4f*         out = (v4f*)d_out;            // [2048*512, 4] f32

    const int n = in_sizes[0] / 4;             // samples (1,048,576)
    const int block = 256;                     // 8 wave32 per block
    const int grid  = (n + block - 1) / block;

    qbottleneck_kernel<<<grid, block, 0, stream>>>(x, prm, out, n);
}